// FGNetTypeB_2920577761788
// MI455X (gfx1250) — compile-verified
//
#include <hip/hip_runtime.h>

typedef __attribute__((ext_vector_type(2))) float v2f;
typedef __attribute__((ext_vector_type(8))) float v8f;

#define MAX_ATOMS 13
#define NUM_IDS   (MAX_ATOMS * MAX_ATOMS)   // 169
#define LATENT    64
#define RANK      128

// ---------------- workspace layout (bytes) ----------------
// counts  : int[169]  @ 0
// cursors : int[169]  @ 1024
// offsets : int[170]  @ 2048
// buckets : int[E]    @ 4096
// ids     : int[E]    @ 4096 + E*4

__global__ void fg_zero_kernel(int* counts, int* cursors) {
    int i = threadIdx.x;
    if (i < NUM_IDS) { counts[i] = 0; cursors[i] = 0; }
}

__global__ void fg_count_kernel(const int* __restrict__ x,
                                const int* __restrict__ fact,
                                int* __restrict__ counts,
                                int* __restrict__ ids, int E) {
    int e = blockIdx.x * blockDim.x + threadIdx.x;
    if (e >= E) return;
    int f0 = fact[e * 3];
    int id = x[f0 * 3 + 1] * MAX_ATOMS + x[f0 * 3 + 2];
    ids[e] = id;
    atomicAdd(&counts[id], 1);
}

__global__ void fg_scan_kernel(const int* __restrict__ counts,
                               int* __restrict__ offsets) {
    if (threadIdx.x == 0) {
        int acc = 0;
        for (int i = 0; i < NUM_IDS; ++i) { offsets[i] = acc; acc += counts[i]; }
        offsets[NUM_IDS] = acc;
    }
}

__global__ void fg_scatter_kernel(const int* __restrict__ ids,
                                  const int* __restrict__ offsets,
                                  int* __restrict__ cursors,
                                  int* __restrict__ buckets, int E) {
    int e = blockIdx.x * blockDim.x + threadIdx.x;
    if (e >= E) return;
    int id = ids[e];
    int pos = atomicAdd(&cursors[id], 1);
    buckets[offsets[id] + pos] = e;
}

// One block.x per atom-pair id; block.y strides over 8-edge tiles of that id's
// edge list. W (64x128 f32 = 32KB) staged once per block in LDS, reused for
// every tile. Each 256-thread block = 8 wave32s; wave w owns the 16-column
// slice [w*16, w*16+16) of the 128-wide output and accumulates K=64 via 16
// chained v_wmma_f32_16x16x4_f32.
__global__ void __launch_bounds__(256)
fg_group_gemm_kernel(const float* __restrict__ nodes,
                     const float* __restrict__ params,
                     const float* __restrict__ bias,
                     const int* __restrict__ fact,
                     const int* __restrict__ offsets,
                     const int* __restrict__ buckets,
                     float* __restrict__ out, int E) {
    const int id    = blockIdx.x;
    const int start = offsets[id];
    const int cnt   = offsets[id + 1] - start;
    if (cnt == 0) return;

    __shared__ float sW[LATENT][RANK + 4];   // stride 132: kills stride-64 bank conflicts, keeps 16B align
    __shared__ float sA[16][LATENT + 4];     // stride 68
    __shared__ float sB[RANK];
    __shared__ int   sEdge[8];

    const int tid = threadIdx.x;

    // Stage W[id] (row-major 64x128) and bias[id] into LDS via float4.
    const float* Wg = params + (size_t)id * LATENT * RANK;
    for (int i = tid; i < (LATENT * RANK) / 4; i += 256) {
        float4 v = ((const float4*)Wg)[i];
        int row = i >> 5;            // (4i)/128
        int col = (i & 31) << 2;     // (4i)%128
        *(float4*)&sW[row][col] = v;
    }
    for (int i = tid; i < RANK; i += 256)
        sB[i] = bias[(size_t)id * RANK + i];

    const int wave  = tid >> 5;          // 0..7 -> column tile
    const int lane  = tid & 31;
    const int m     = lane & 15;         // A row / C col-index within tile
    const int khalf = (lane >> 4) << 1;  // 0 for lanes 0-15, 2 for lanes 16-31
    const int ncol  = (wave << 4) + m;   // global output column 0..127

    const int numTiles = (cnt + 7) >> 3;
    for (int t = blockIdx.y; t < numTiles; t += gridDim.y) {
        __syncthreads();   // protect sA/sEdge reuse (and first-iter sW visibility)
        if (tid < 8) {
            int ei = (t << 3) + tid;
            sEdge[tid] = (ei < cnt) ? buckets[start + ei] : -1;
        }
        __syncthreads();

        // Gather A: 16 rows x 64 f32. thread -> (row = tid/16, 4 cols).
        {
            int row  = tid >> 4;
            int col  = (tid & 15) << 2;
            int e    = sEdge[row >> 1];
            float4 av = make_float4(0.f, 0.f, 0.f, 0.f);
            if (e >= 0) {
                int node = fact[e * 3 + 1 + (row & 1)];   // sub_fact[e][row%2]
                av = *(const float4*)(nodes + (size_t)node * LATENT + col);
            }
            *(float4*)&sA[row][col] = av;
        }
        __syncthreads();

        // 16 x (16x16x4 f32 WMMA), accumulator chained through C.
        v8f acc = {};
        #pragma unroll
        for (int k = 0; k < LATENT; k += 4) {
            v2f a, b;
            a.x = sA[m][k + khalf];
            a.y = sA[m][k + khalf + 1];
            b.x = sW[k + khalf][ncol];
            b.y = sW[k + khalf + 1][ncol];
            acc = __builtin_amdgcn_wmma_f32_16x16x4_f32(
                false, a, false, b, (short)0, acc, false, false);
        }

        // Epilogue: bias + ReLU, scatter to out[k, e, r] = out[(r&1)*E*128 + e*128 + col].
        float bv = sB[ncol];
        #pragma unroll
        for (int v = 0; v < 8; ++v) {
            int crow = v + ((lane >> 4) << 3);   // C-matrix M row 0..15
            int e    = sEdge[crow >> 1];
            if (e >= 0) {
                float val = acc[v] + bv;
                val = val > 0.f ? val : 0.f;
                out[(size_t)(crow & 1) * E * RANK + (size_t)e * RANK + ncol] = val;
            }
        }
    }
}

extern "C" void kernel_launch(void* const* d_in, const int* in_sizes, int n_in,
                              void* d_out, int out_size, void* d_ws, size_t ws_size,
                              hipStream_t stream) {
    const float* nodes  = (const float*)d_in[0];
    const float* params = (const float*)d_in[1];
    const float* bias   = (const float*)d_in[2];
    const int*   x      = (const int*)d_in[3];
    const int*   fact   = (const int*)d_in[4];
    const int E = in_sizes[4] / 3;

    char* ws = (char*)d_ws;
    int* counts  = (int*)(ws + 0);
    int* cursors = (int*)(ws + 1024);
    int* offsets = (int*)(ws + 2048);
    int* buckets = (int*)(ws + 4096);
    int* ids     = (int*)(ws + 4096 + (size_t)E * sizeof(int));

    float* out = (float*)d_out;

    fg_zero_kernel<<<1, 256, 0, stream>>>(counts, cursors);
    fg_count_kernel<<<(E + 255) / 256, 256, 0, stream>>>(x, fact, counts, ids, E);
    fg_scan_kernel<<<1, 32, 0, stream>>>(counts, offsets);
    fg_scatter_kernel<<<(E + 255) / 256, 256, 0, stream>>>(ids, offsets, cursors, buckets, E);

    dim3 grid(NUM_IDS, 8);   // 169 groups x 8 tile-strided blocks = 1352 blocks
    fg_group_gemm_kernel<<<grid, 256, 0, stream>>>(nodes, params, bias, fact,
                                                   offsets, buckets, out, E);
}